// myPCANorm_noRec_76836964925865
// MI455X (gfx1250) — compile-verified
//
#include <hip/hip_runtime.h>
#include <hip/hip_bf16.h>
#include <cstdint>

typedef float v2f __attribute__((ext_vector_type(2)));
typedef float v8f __attribute__((ext_vector_type(8)));
typedef int   vi4 __attribute__((vector_size(16)));   // matches async-LDS builtin pointee

#define EPSV   1e-5f
#define CH     64
#define HW     3136          // 56*56
#define MTOT   401408        // 128*3136
#define NPOW   10
#define NEIG   64
#define NBLK1  256

// workspace layout (float offsets)
#define OFF_PG 0u                        // partial Grams: NBLK1 * 4096
#define OFF_PS (NBLK1 * 4096u)           // partial sums:  NBLK1 * 64
#define OFF_G  (OFF_PS + NBLK1 * 64u)    // final Gram:    4096
#define OFF_S  (OFF_G + 4096u)           // final sums:    64
#define OFF_A  (OFF_S + 64u)             // Aeff:          4096
#define OFF_B  (OFF_A + 4096u)           // beff:          64

// ---- optional CDNA5 async LDS path (GLOBAL_LOAD_ASYNC_TO_LDS_B128) --------
#if defined(__has_builtin)
#if __has_builtin(__builtin_amdgcn_global_load_async_to_lds_b128)
#define HAVE_ASYNC_LDS 1
#endif
#endif
#ifndef HAVE_ASYNC_LDS
#define HAVE_ASYNC_LDS 0
#endif

// raw SOPP waits (real gfx1250 opcodes; avoids builtin-name uncertainty)
#define WAIT_ASYNCCNT_8() asm volatile("s_wait_asynccnt 0x8" ::: "memory")
#define WAIT_ASYNCCNT_0() asm volatile("s_wait_asynccnt 0x0" ::: "memory")
#define WAIT_DSCNT_0()    asm volatile("s_wait_dscnt 0x0"    ::: "memory")

// Stage a 64ch x 16col tile into an LDS buffer (padded row stride 20 floats).
// Async path: 8 GLOBAL_LOAD_ASYNC_TO_LDS_B128 per lane (ASYNCcnt-tracked, no
// VGPR round-trip). Fallback: coalesced b128 load + ds_store_b128.
__device__ __forceinline__ void stage_tile(const float* __restrict__ px,
                                           float* buf, int lane) {
#pragma unroll
  for (int i = 0; i < 8; i++) {
    const int c = i * 8 + (lane >> 2);
    const int q = (lane & 3) * 4;
#if HAVE_ASYNC_LDS
    __builtin_amdgcn_global_load_async_to_lds_b128(
        (__attribute__((address_space(1))) vi4*)(px + (size_t)c * HW + q),
        (__attribute__((address_space(3))) vi4*)(buf + c * 20 + q), 0, 0);
#else
    const float4 v = *(const float4*)(px + (size_t)c * HW + q);
    *(float4*)(buf + c * 20 + q) = v;
#endif
  }
}

__device__ __forceinline__ const float* tile_ptr(const float* __restrict__ x,
                                                 int tb, int wave) {
  const int sub = tb * 8 + wave;        // 16-col subtile id in [0, 25088)
  const int n   = sub / 196;
  const int hw0 = (sub % 196) * 16;
  return x + ((size_t)n * CH) * HW + hw0;
}

// ---------------------------------------------------------------------------
// Kernel 1: partial Gram G = X * X^T (64x64) + per-channel sums via
// V_WMMA_F32_16X16X4_F32, double-buffered LDS staging. For the symmetric Gram,
// the A-fragment of channel-tile i and B-fragment of channel-tile j have
// identical lane layouts, so one fragment set feeds both operands; a fifth
// WMMA against an all-ones B fragment yields per-channel column sums free.
// ---------------------------------------------------------------------------
__global__ __launch_bounds__(256) void gram_kernel(const float* __restrict__ x,
                                                   float* __restrict__ ws) {
  __shared__ float smem[8 * 2 * 64 * 20];   // 8 waves x 2 buffers x 64x20 = 80KB
  const int tid  = threadIdx.x;
  const int lane = tid & 31;
  const int wave = tid >> 5;
  const int half = lane >> 4;
  const int l    = lane & 15;
  float* buf0 = smem + wave * (2 * 64 * 20);
  float* buf1 = buf0 + 64 * 20;

  v8f acc[4][4];
  v8f accS[4];
#pragma unroll
  for (int i = 0; i < 4; i++) {
#pragma unroll
    for (int j = 0; j < 4; j++)
#pragma unroll
      for (int r = 0; r < 8; r++) acc[i][j][r] = 0.0f;
#pragma unroll
    for (int r = 0; r < 8; r++) accS[i][r] = 0.0f;
  }
  const v2f ones = {1.0f, 1.0f};

  // prologue: stage first tile
  stage_tile(tile_ptr(x, blockIdx.x, wave), buf0, lane);

  int parity = 0;
  for (int tb = blockIdx.x; tb < 3136; tb += gridDim.x) {
    float* cur = parity ? buf1 : buf0;
    float* nxt = parity ? buf0 : buf1;
    const int tbn = tb + (int)gridDim.x;
    if (tbn < 3136) {
#if HAVE_ASYNC_LDS
      WAIT_DSCNT_0();                  // prior ds reads of 'nxt' are done
#endif
      stage_tile(tile_ptr(x, tbn, wave), nxt, lane);
#if HAVE_ASYNC_LDS
      WAIT_ASYNCCNT_8();               // in-order: current tile's 8 are complete
#endif
    } else {
#if HAVE_ASYNC_LDS
      WAIT_ASYNCCNT_0();
#endif
    }

#pragma unroll
    for (int s = 0; s < 4; s++) {
      const int m4 = s * 4 + 2 * half;
      v2f frag[4];
#pragma unroll
      for (int ct = 0; ct < 4; ct++)
        frag[ct] = *(const v2f*)(cur + (ct * 16 + l) * 20 + m4);
#pragma unroll
      for (int i = 0; i < 4; i++) {
#pragma unroll
        for (int j = 0; j < 4; j++)
          acc[i][j] = __builtin_amdgcn_wmma_f32_16x16x4_f32(
              false, frag[i], false, frag[j], (short)0, acc[i][j], false, false);
        accS[i] = __builtin_amdgcn_wmma_f32_16x16x4_f32(
            false, frag[i], false, ones, (short)0, accS[i], false, false);
      }
    }
    parity ^= 1;
  }

  // deterministic cross-wave reduction of partial Gram / sums through LDS
  __syncthreads();
  float* stage = smem;                     // reuse staging memory (8*256 floats)
  const int blk = blockIdx.x;
#pragma unroll
  for (int i = 0; i < 4; i++) {
#pragma unroll
    for (int j = 0; j < 4; j++) {
#pragma unroll
      for (int r = 0; r < 8; r++) stage[wave * 256 + r * 32 + lane] = acc[i][j][r];
      __syncthreads();
      float v = 0.0f;
#pragma unroll
      for (int w = 0; w < 8; w++) v += stage[w * 256 + tid];
      const int r   = tid >> 5;
      const int L   = tid & 31;
      const int row = i * 16 + r + ((L >= 16) ? 8 : 0);
      const int col = j * 16 + (L & 15);
      ws[OFF_PG + (size_t)blk * 4096 + row * 64 + col] = v;
      __syncthreads();
    }
  }
#pragma unroll
  for (int i = 0; i < 4; i++) {
#pragma unroll
    for (int r = 0; r < 8; r++) stage[wave * 256 + r * 32 + lane] = accS[i][r];
    __syncthreads();
    if ((tid & 15) == 0) {                 // columns of the ones-tile are identical
      float v = 0.0f;
#pragma unroll
      for (int w = 0; w < 8; w++) v += stage[w * 256 + tid];
      const int r   = tid >> 5;
      const int L   = tid & 31;
      const int row = i * 16 + r + ((L >= 16) ? 8 : 0);
      ws[OFF_PS + (size_t)blk * 64 + row] = v;
    }
    __syncthreads();
  }
}

// ---------------------------------------------------------------------------
// Kernel 2: deterministic reduction of the NBLK1 partials.
// ---------------------------------------------------------------------------
__global__ __launch_bounds__(256) void reduce_kernel(float* __restrict__ ws) {
  const int idx = blockIdx.x * 256 + threadIdx.x;
  if (idx < 4096) {
    float a = 0.0f;
    for (int b = 0; b < NBLK1; b++) a += ws[OFF_PG + (size_t)b * 4096 + idx];
    ws[OFF_G + idx] = a;
  } else if (idx < 4160) {
    const int c = idx - 4096;
    float a = 0.0f;
    for (int b = 0; b < NBLK1; b++) a += ws[OFF_PS + (size_t)b * 64 + c];
    ws[OFF_S + c] = a;
  }
}

// ---------------------------------------------------------------------------
// Kernel 3: form cov from Gram + sums, run 64 deflated power iterations
// (serial chain per the reference), emit fused Aeff / beff.
// ---------------------------------------------------------------------------
__global__ __launch_bounds__(64) void eigen_kernel(float* __restrict__ ws,
                                                   const float* __restrict__ weight,
                                                   const float* __restrict__ bias) {
  __shared__ float cov[64][65];
  __shared__ float mu_s[64], is_s[64], vbuf[64], red[64];
  const int c = threadIdx.x;
  const float Mf  = (float)MTOT;
  const float Mm1 = (float)(MTOT - 1);

  const float m   = ws[OFF_S + c] / Mf;
  const float var = (ws[OFF_G + c * 64 + c] - Mf * m * m) / Mm1;   // unbiased
  const float inv = 1.0f / sqrtf(var + EPSV);
  mu_s[c] = m;
  is_s[c] = inv;
  __syncthreads();

  for (int j = 0; j < 64; j++) {
    float val = (ws[OFF_G + c * 64 + j] - Mf * m * mu_s[j]) * inv * is_s[j] / Mm1;
    if (j == c) val += EPSV;
    cov[c][j] = val;
  }
  __syncthreads();

  for (int e = 0; e < NEIG; e++) {
    vbuf[c] = 1.0f;
    __syncthreads();
    for (int it = 0; it < NPOW; it++) {
      float w = 0.0f;
      for (int k = 0; k < 64; k++) w = fmaf(cov[c][k], vbuf[k], w);
      red[c] = w * w;
      __syncthreads();
      for (int off = 32; off > 0; off >>= 1) {
        if (c < off) red[c] += red[c + off];
        __syncthreads();
      }
      const float den = fmaxf(sqrtf(red[0]), EPSV);
      __syncthreads();
      vbuf[c] = w / den;
      __syncthreads();
    }
    // deflation: cov -= outer(cov @ v, v); each thread owns row c
    float w2 = 0.0f;
    for (int k = 0; k < 64; k++) w2 = fmaf(cov[c][k], vbuf[k], w2);
    for (int j = 0; j < 64; j++) cov[c][j] -= w2 * vbuf[j];

    // fused affine: Aeff[e,k] = weight[e]*v[k]*inv_s[k]; beff[e] = bias[e]-Aeff[e,:]@mu
    const float a = weight[e] * vbuf[c] * is_s[c];
    ws[OFF_A + e * 64 + c] = a;
    red[c] = a * mu_s[c];
    __syncthreads();
    for (int off = 32; off > 0; off >>= 1) {
      if (c < off) red[c] += red[c + off];
      __syncthreads();
    }
    if (c == 0) ws[OFF_B + e] = bias[e] - red[0];
    __syncthreads();
  }
}

// ---------------------------------------------------------------------------
// Kernel 4: out[c,m] = sum_k Aeff[c,k]*x[k,m] + beff[c] via WMMA f32 16x16x4.
// B fragments (N = m columns) load directly coalesced from global; A fragments
// come from LDS-staged Aeff (padded stride 66 -> conflict-free). 4 tiles per
// wave amortize the Aeff staging.
// ---------------------------------------------------------------------------
__global__ __launch_bounds__(256) void apply_kernel(const float* __restrict__ x,
                                                    const float* __restrict__ ws,
                                                    float* __restrict__ out) {
  __shared__ float lA[64 * 66];
  __shared__ float lB[64];
  const int tid  = threadIdx.x;
  const int lane = tid & 31;
  const int wave = tid >> 5;
  const int half = lane >> 4;
  const int l    = lane & 15;

#pragma unroll
  for (int i = 0; i < 16; i++) {
    const int idx = i * 256 + tid;
    lA[(idx >> 6) * 66 + (idx & 63)] = ws[OFF_A + idx];
  }
  if (tid < 64) lB[tid] = ws[OFF_B + tid];
  __syncthreads();

  for (int t = 0; t < 4; t++) {
    const int sub = blockIdx.x * 32 + t * 8 + wave;   // 784*32 = 25088 tiles
    const int n   = sub / 196;
    const int hw0 = (sub % 196) * 16;
    const float* px = x + ((size_t)n * CH) * HW + hw0;

    v8f acc[4];
#pragma unroll
    for (int ct = 0; ct < 4; ct++)
#pragma unroll
      for (int r = 0; r < 8; r++) acc[ct][r] = 0.0f;

#pragma unroll
    for (int s = 0; s < 16; s++) {
      const int k0   = s * 4;
      const int row0 = k0 + 2 * half;
      v2f b;
      b.x = px[(size_t)row0 * HW + l];
      b.y = px[(size_t)(row0 + 1) * HW + l];
#pragma unroll
      for (int ct = 0; ct < 4; ct++) {
        const v2f a = *(const v2f*)(lA + (ct * 16 + l) * 66 + k0 + 2 * half);
        acc[ct] = __builtin_amdgcn_wmma_f32_16x16x4_f32(
            false, a, false, b, (short)0, acc[ct], false, false);
      }
    }

    float* po = out + ((size_t)n * CH) * HW + hw0;
#pragma unroll
    for (int ct = 0; ct < 4; ct++)
#pragma unroll
      for (int r = 0; r < 8; r++) {
        const int c = ct * 16 + r + 8 * half;
        po[(size_t)c * HW + l] = acc[ct][r] + lB[c];
      }
  }
}

// ---------------------------------------------------------------------------
extern "C" void kernel_launch(void* const* d_in, const int* in_sizes, int n_in,
                              void* d_out, int out_size, void* d_ws, size_t ws_size,
                              hipStream_t stream) {
  (void)in_sizes; (void)n_in; (void)out_size; (void)ws_size;
  const float* x      = (const float*)d_in[0];
  const float* weight = (const float*)d_in[1];
  const float* bias   = (const float*)d_in[2];
  float* out = (float*)d_out;
  float* ws  = (float*)d_ws;

  gram_kernel<<<NBLK1, 256, 0, stream>>>(x, ws);
  reduce_kernel<<<17, 256, 0, stream>>>(ws);
  eigen_kernel<<<1, 64, 0, stream>>>(ws, weight, bias);
  apply_kernel<<<784, 256, 0, stream>>>(x, ws, out);
}